// GATEncoder_42769284334212
// MI455X (gfx1250) — compile-verified
//
#include <hip/hip_runtime.h>
#include <hip/hip_bf16.h>

typedef float v2f __attribute__((ext_vector_type(2)));
typedef float v8f __attribute__((ext_vector_type(8)));

#define D1    256   // HEADS*HID
#define HEADS 4
#define HID   64

// ---------------------------------------------------------------------------
// GEMM: C[M,256] = A[M,K] @ B[K,256] using V_WMMA_F32_16X16X4_F32.
// One wave computes one 16x16 tile of C, stepping K by 4.
// A operand layout (ISA 7.12.2, 32-bit A 16x4): lanes 0-15 hold {K=k0,k0+1}
// for row M=lane; lanes 16-31 hold {K=k0+2,k0+3} for row M=lane-16.
// B operand mirrored: v.x = B[k0+2*(lane>>4)][n0+lane&15], v.y = next K row.
// C/D layout: VGPR i -> row i (lanes 0-15) / row 8+i (lanes 16-31), col lane&15.
// ---------------------------------------------------------------------------
__global__ void gemm_wmma_f32(const float* __restrict__ A,
                              const float* __restrict__ B,
                              float* __restrict__ C,
                              int M, int K) {
    const int NT = D1 / 16;                      // 16 tiles across N
    int wave  = (blockIdx.x * blockDim.x + threadIdx.x) >> 5;
    int lane  = threadIdx.x & 31;
    int tileN = wave % NT;
    int tileM = wave / NT;
    int m0 = tileM * 16;
    int n0 = tileN * 16;
    if (m0 >= M) return;

    int mrow  = lane & 15;
    int khalf = lane >> 4;                        // 0 or 1
    int arow  = m0 + mrow;
    if (arow >= M) arow = M - 1;                  // clamp (M is a multiple of 16 here)

    v8f acc = {};
    for (int k0 = 0; k0 < K; k0 += 4) {
        const float* ap = A + (size_t)arow * K + (k0 + khalf * 2);
        v2f a; a.x = ap[0]; a.y = ap[1];
        const float* bp = B + (size_t)(k0 + khalf * 2) * D1 + n0 + mrow;
        v2f b; b.x = bp[0]; b.y = bp[D1];
        acc = __builtin_amdgcn_wmma_f32_16x16x4_f32(
            /*neg_a=*/false, a, /*neg_b=*/false, b,
            /*c_mod=*/(short)0, acc, /*reuse_a=*/false, /*reuse_b=*/false);
    }

    int rbase = m0 + khalf * 8;
    float* cp = C + (size_t)rbase * D1 + n0 + mrow;
#pragma unroll
    for (int i = 0; i < 8; ++i) {
        if (rbase + i < M) cp[(size_t)i * D1] = acc[i];
    }
}

// ---------------------------------------------------------------------------
// Per-node attention logits: al[n,h] = sum_c h[n,h*HID+c] * a[h,c]
// ---------------------------------------------------------------------------
__global__ void logits_kernel(const float* __restrict__ h,
                              const float* __restrict__ a_src,
                              const float* __restrict__ a_dst,
                              float* __restrict__ al_src,
                              float* __restrict__ al_dst, int n) {
    int idx = blockIdx.x * blockDim.x + threadIdx.x;   // (node, head)
    if (idx >= n * HEADS) return;
    int node = idx >> 2, hd = idx & (HEADS - 1);
    const float* hp = h + (size_t)node * D1 + hd * HID;
    const float* as = a_src + hd * HID;
    const float* ad = a_dst + hd * HID;
    float s1 = 0.f, s2 = 0.f;
#pragma unroll 8
    for (int c = 0; c < HID; ++c) {
        float v = hp[c];
        s1 += v * as[c];
        s2 += v * ad[c];
    }
    al_src[idx] = s1;
    al_dst[idx] = s2;
}

__device__ __forceinline__ void edge_sd(const long long* __restrict__ ei,
                                        int ne, int e, int& s, int& d) {
    if (e < ne) { s = (int)ei[e]; d = (int)ei[ne + e]; }
    else        { s = d = e - ne; }                       // self loop
}

// order-preserving float->uint map for atomicMax-based segment max
__device__ __forceinline__ unsigned ford(float f) {
    unsigned b = __float_as_uint(f);
    return (b & 0x80000000u) ? ~b : (b | 0x80000000u);
}
__device__ __forceinline__ float funord(unsigned o) {
    unsigned b = (o & 0x80000000u) ? (o & 0x7FFFFFFFu) : ~o;
    return __uint_as_float(b);
}

// pass 1: raw scores + leaky relu + segment max over dst
__global__ void edge_score_max(const long long* __restrict__ ei, int ne, int etot,
                               const float* __restrict__ al_src,
                               const float* __restrict__ al_dst,
                               float* __restrict__ escore,
                               unsigned* __restrict__ mmax) {
    int e = blockIdx.x * blockDim.x + threadIdx.x;
    if (e >= etot) return;
    int s, d; edge_sd(ei, ne, e, s, d);
#pragma unroll
    for (int h = 0; h < HEADS; ++h) {
        float v = al_src[s * HEADS + h] + al_dst[d * HEADS + h];
        v = (v > 0.f) ? v : 0.2f * v;                  // leaky relu, slope 0.2
        escore[(size_t)e * HEADS + h] = v;
        atomicMax(&mmax[d * HEADS + h], ford(v));
    }
}

// pass 2: exp(e - m[dst]) + segment sum over dst (escore overwritten with exp)
__global__ void edge_exp_sum(const long long* __restrict__ ei, int ne, int etot,
                             float* __restrict__ escore,
                             const unsigned* __restrict__ mmax,
                             float* __restrict__ ssum) {
    int e = blockIdx.x * blockDim.x + threadIdx.x;
    if (e >= etot) return;
    int s, d; edge_sd(ei, ne, e, s, d);
    (void)s;
#pragma unroll
    for (int h = 0; h < HEADS; ++h) {
        float m  = funord(mmax[d * HEADS + h]);
        float ex = expf(escore[(size_t)e * HEADS + h] - m);
        escore[(size_t)e * HEADS + h] = ex;
        atomicAdd(&ssum[d * HEADS + h], ex);
    }
}

// pass 3: agg[dst] += alpha * h[src]  -- one wave per edge, 8 channels/lane
__global__ void edge_aggregate(const long long* __restrict__ ei, int ne, int etot,
                               const float* __restrict__ escore,
                               const float* __restrict__ ssum,
                               const float* __restrict__ hlin,
                               float* __restrict__ agg) {
    int e = blockIdx.x * (blockDim.x >> 5) + (threadIdx.x >> 5);
    if (e >= etot) return;
    int lane = threadIdx.x & 31;
    int s, d; edge_sd(ei, ne, e, s, d);
    int hd = lane >> 3;                                  // 8 lanes per head
    float alpha = escore[(size_t)e * HEADS + hd] /
                  (ssum[d * HEADS + hd] + 1e-16f);
    const float* hp = hlin + (size_t)s * D1 + lane * 8;
    float*       op = agg  + (size_t)d * D1 + lane * 8;
    float4 v0 = *(const float4*)(hp);
    float4 v1 = *(const float4*)(hp + 4);
    atomicAdd(&op[0], alpha * v0.x);
    atomicAdd(&op[1], alpha * v0.y);
    atomicAdd(&op[2], alpha * v0.z);
    atomicAdd(&op[3], alpha * v0.w);
    atomicAdd(&op[4], alpha * v1.x);
    atomicAdd(&op[5], alpha * v1.y);
    atomicAdd(&op[6], alpha * v1.z);
    atomicAdd(&op[7], alpha * v1.w);
}

// out = relu(in + bias) ; safe in-place
__global__ void bias_relu(const float* __restrict__ in,
                          const float* __restrict__ bias,
                          float* __restrict__ out, int total) {
    int i = blockIdx.x * blockDim.x + threadIdx.x;
    if (i >= total) return;
    float v = in[i] + bias[i & (D1 - 1)];
    out[i] = v > 0.f ? v : 0.f;
}

// ---------------------------------------------------------------------------
extern "C" void kernel_launch(void* const* d_in, const int* in_sizes, int n_in,
                              void* d_out, int out_size, void* d_ws, size_t ws_size,
                              hipStream_t stream) {
    const float*     x   = (const float*)d_in[0];
    const long long* ei  = (const long long*)d_in[1];   // int64 edge_index [2,E]
    const float*     W1  = (const float*)d_in[2];
    const float*     a1s = (const float*)d_in[3];
    const float*     a1d = (const float*)d_in[4];
    const float*     b1  = (const float*)d_in[5];
    const float*     W2  = (const float*)d_in[6];
    const float*     a2s = (const float*)d_in[7];
    const float*     a2d = (const float*)d_in[8];
    const float*     b2  = (const float*)d_in[9];
    float* out = (float*)d_out;

    const int d1   = in_sizes[5];            // 256
    const int K1   = in_sizes[2] / d1;       // 128
    const int n    = in_sizes[0] / K1;       // 50000
    const int ne   = in_sizes[1] / 2;        // 500000
    const int etot = ne + n;                 // + self loops

    // workspace partition
    float*    h_lin  = (float*)d_ws;                        // n*256
    float*    x2     = h_lin + (size_t)n * D1;              // n*256
    float*    al_s   = x2 + (size_t)n * D1;                 // n*4
    float*    al_d   = al_s + (size_t)n * HEADS;            // n*4
    unsigned* mmax   = (unsigned*)(al_d + (size_t)n * HEADS); // n*4
    float*    ssum   = (float*)(mmax + (size_t)n * HEADS);  // n*4
    float*    escore = ssum + (size_t)n * HEADS;            // etot*4

    const int BT = 256;
    int gemm_blocks   = (((n + 15) / 16) * (D1 / 16) * 32 + BT - 1) / BT;
    int logit_blocks  = (n * HEADS + BT - 1) / BT;
    int edge_blocks   = (etot + BT - 1) / BT;
    int agg_blocks    = (etot + (BT / 32) - 1) / (BT / 32);
    int elem_blocks   = (n * D1 + BT - 1) / BT;

    for (int layer = 0; layer < 2; ++layer) {
        const float* xin = (layer == 0) ? x   : x2;
        const float* W   = (layer == 0) ? W1  : W2;
        const float* as  = (layer == 0) ? a1s : a2s;
        const float* ad  = (layer == 0) ? a1d : a2d;
        const float* bb  = (layer == 0) ? b1  : b2;
        const int    K   = (layer == 0) ? K1  : d1;
        float*       agg = (layer == 0) ? x2  : out;

        hipMemsetAsync(mmax, 0, (size_t)n * HEADS * sizeof(unsigned), stream); // ford identity
        hipMemsetAsync(ssum, 0, (size_t)n * HEADS * sizeof(float), stream);
        hipMemsetAsync(agg,  0, (size_t)n * D1 * sizeof(float), stream);

        gemm_wmma_f32<<<gemm_blocks, BT, 0, stream>>>(xin, W, h_lin, n, K);
        logits_kernel<<<logit_blocks, BT, 0, stream>>>(h_lin, as, ad, al_s, al_d, n);
        edge_score_max<<<edge_blocks, BT, 0, stream>>>(ei, ne, etot, al_s, al_d, escore, mmax);
        edge_exp_sum<<<edge_blocks, BT, 0, stream>>>(ei, ne, etot, escore, mmax, ssum);
        edge_aggregate<<<agg_blocks, BT, 0, stream>>>(ei, ne, etot, escore, ssum, h_lin, agg);
        bias_relu<<<elem_blocks, BT, 0, stream>>>(agg, bb, agg, n * D1);
    }
}